// MaskMLP_69303592288302
// MI455X (gfx1250) — compile-verified
//
#include <hip/hip_runtime.h>
#include <math.h>

// ---------- problem constants (from reference) ----------
#define LS        512          // last dim of x
#define HID       256          // HIDDEN
#define SMP       4            // SAMPLING
#define SEG       128          // LS / SMP
#define THREADS   256          // 8 waves
#define SLAB_ROWS 256          // rows per block
#define CHUNK     32           // rows per inner iteration (2 tiles of 16)
#define N_CHUNKS  (SLAB_ROWS / CHUNK)

typedef float v2f __attribute__((ext_vector_type(2)));
typedef float v4f __attribute__((ext_vector_type(4)));
typedef float v8f __attribute__((ext_vector_type(8)));
typedef unsigned int v4u __attribute__((ext_vector_type(4)));
typedef int v4i __attribute__((ext_vector_type(4)));
typedef int v8i __attribute__((ext_vector_type(8)));

// LDS strides in dwords (padded for bank-conflict-free b64 fragment reads)
#define W1_STRIDE 132   // 128 + 4
#define W2_STRIDE 260   // 256 + 4
#define H_STRIDE  260   // 256 + 4

#define LDS_W1_DW (256 * W1_STRIDE)   // 33792 dw
#define LDS_W2_DW (128 * W2_STRIDE)   // 33280 dw
#define LDS_H_DW  (CHUNK * H_STRIDE)  //  8320 dw  -> total 301,568 B

#if defined(__has_builtin)
#  if __has_builtin(__builtin_amdgcn_tensor_load_to_lds)
#    define USE_TDM 1
#  endif
#endif
#ifndef USE_TDM
#  define USE_TDM 0
#endif

__device__ __forceinline__ float gelu_exact(float t) {
    return 0.5f * t * (1.0f + erff(t * 0.7071067811865476f));
}

// ------------------------------------------------------------------
// Pre-pass: pack masked/strided weights contiguously into workspace.
//   w1p[g][n][k]  = wes1[n, 4k+g]          (4*256*128)
//   w2p[g][n][k]  = wes2[128g+n, 256g+k]   (4*128*256)
//   w3d[i][t]     = wes3[i, 4i+t]          (128*4)
// ------------------------------------------------------------------
__global__ void maskmlp_pack_kernel(const float* __restrict__ wes1,
                                    const float* __restrict__ wes2,
                                    const float* __restrict__ wes3,
                                    float* __restrict__ w1p,
                                    float* __restrict__ w2p,
                                    float* __restrict__ w3d) {
    int idx = blockIdx.x * blockDim.x + threadIdx.x;
    if (idx < 4 * 256 * 128) {
        int g = idx >> 15;
        int r = idx & 32767;
        int n = r >> 7, k = r & 127;
        w1p[idx] = wes1[n * LS + 4 * k + g];
    } else if (idx < 2 * 4 * 256 * 128) {
        int j = idx - 4 * 256 * 128;
        int g = j >> 15;
        int r = j & 32767;
        int n = r >> 8, k = r & 255;
        w2p[j] = wes2[(g * SEG + n) * (SMP * HID) + g * HID + k];
    } else if (idx < 2 * 4 * 256 * 128 + SEG * SMP) {
        int j = idx - 2 * 4 * 256 * 128;
        int i = j >> 2, t = j & 3;
        w3d[j] = wes3[i * LS + 4 * i + t];
    }
}

#if USE_TDM
// Issue one TDM 2D tile load: rows x rowDw dwords, LDS-padded per D# pad fields.
// pad_interval encodes 2^(v+1) dwords between pads, pad_amount = (v+1) dwords.
__device__ __forceinline__ void tdm_load_2d(const float* gsrc, float* ldst,
                                            unsigned rowDw, unsigned rows,
                                            unsigned pad_interval, unsigned pad_amount) {
    unsigned long long ga = (unsigned long long)(const void*)gsrc;
    unsigned ldsOff = (unsigned)(unsigned long long)(void*)ldst;  // flat->LDS: low 32 bits
    v4u g0;
    g0.x = 1u;                                                    // count=1, user mode
    g0.y = ldsOff;                                                // lds_addr
    g0.z = (unsigned)ga;                                          // global_addr[31:0]
    g0.w = ((unsigned)(ga >> 32) & 0x1FFFFFFu) | (2u << 30);      // addr[56:32] | type=2
    v8i g1;
    g1[0] = (int)((2u << 16) | (1u << 20) |                       // data_size=4B, pad_en
                  (pad_interval << 22) | (pad_amount << 25));
    g1[1] = (int)(rowDw << 16);                                   // tensor_dim0 lo16
    g1[2] = (int)(rows  << 16);                                   // dim0 hi=0 | tensor_dim1 lo16
    g1[3] = (int)(rowDw << 16);                                   // dim1 hi=0 | tile_dim0
    g1[4] = (int)rows;                                            // tile_dim1 | tile_dim2=0
    g1[5] = (int)rowDw;                                           // tensor_dim0_stride lo32
    g1[6] = 0;
    g1[7] = 0;
    v4i z4 = {0, 0, 0, 0};
#if defined(__clang_major__) && (__clang_major__ >= 23)
    v8i z8 = {0, 0, 0, 0, 0, 0, 0, 0};
    __builtin_amdgcn_tensor_load_to_lds(g0, g1, z4, z4, z8, 0);
#else
    __builtin_amdgcn_tensor_load_to_lds(g0, g1, z4, z4, 0);
#endif
}
#endif

// ------------------------------------------------------------------
// Fused MaskMLP. Block = (group g, 256-row slab). Weights staged once
// (TDM DMA -> LDS), then 8 chunks of 32 rows run both GEMMs from LDS.
//   wave = 4*t2 + nq : t2 = 16-row tile in chunk, nq = N quarter
// ------------------------------------------------------------------
__global__ void __launch_bounds__(THREADS)
maskmlp_main_kernel(const float* __restrict__ x,
                    const float* __restrict__ bias1,
                    const float* __restrict__ bias2,
                    const float* __restrict__ bias3,
                    const float* __restrict__ w1p,
                    const float* __restrict__ w2p,
                    const float* __restrict__ w3d,
                    float* __restrict__ out) {
    __shared__ float lds_w1[LDS_W1_DW];
    __shared__ float lds_w2[LDS_W2_DW];
    __shared__ float lds_h[LDS_H_DW];

    const int tid  = threadIdx.x;
    const int lane = tid & 31;
    const int wave = tid >> 5;          // 0..7
    const int t2   = wave >> 2;         // 0..1  tile within 32-row chunk
    const int nq   = wave & 3;          // 0..3  N quarter
    const int lm   = lane & 15;
    const int hi   = lane >> 4;         // K sub-offset = 2*hi

    const int g = blockIdx.x & 3;
    const long row0 = (long)(blockIdx.x >> 2) * SLAB_ROWS;

    // -------- stage W1_g (256x128) and W2_g (128x256) once, via TDM --------
#if USE_TDM
    if (wave == 0) {
        tdm_load_2d(w1p + (size_t)g * 256 * 128, lds_w1, 128, 256, 6, 3); // pad 4dw / 128dw
        __builtin_amdgcn_s_wait_tensorcnt(0);
    } else if (wave == 1) {
        tdm_load_2d(w2p + (size_t)g * 128 * 256, lds_w2, 256, 128, 7, 3); // pad 4dw / 256dw
        __builtin_amdgcn_s_wait_tensorcnt(0);
    }
#else
    {
        const v4f* s1 = (const v4f*)(w1p + (size_t)g * 256 * 128);
        for (int e = tid; e < 256 * 128 / 4; e += THREADS) {
            int n = e >> 5, k4 = (e & 31) << 2;
            *(v4f*)&lds_w1[n * W1_STRIDE + k4] = s1[e];
        }
        const v4f* s2 = (const v4f*)(w2p + (size_t)g * 128 * 256);
        for (int e = tid; e < 128 * 256 / 4; e += THREADS) {
            int n = e >> 6, k4 = (e & 63) << 2;
            *(v4f*)&lds_w2[n * W2_STRIDE + k4] = s2[e];
        }
    }
#endif
    __syncthreads();

    for (int ch = 0; ch < N_CHUNKS; ++ch) {
        const long tileRow = row0 + ch * CHUNK + t2 * 16;
        const float* xrow = x + (tileRow + lm) * LS;

        // ---- A fragments of X_g (16 rows x K=128): lane<16 K{0,1}, lane>=16 K{2,3} ----
        v2f a1[32];
        #pragma unroll
        for (int kk = 0; kk < 32; ++kk) {
            int c0 = 16 * kk + 8 * hi + g;       // x column of K = 4*kk + 2*hi
            a1[kk][0] = xrow[c0];
            a1[kk][1] = xrow[c0 + 4];
        }

        // ---- layer 1: H_g(16x64 per wave) = X_g @ W1_g^T, +b1, gelu -> LDS ----
        #pragma unroll
        for (int nt = 0; nt < 4; ++nt) {
            const int nloc = nq * 64 + nt * 16;
            const int brow = (nloc + lm) * W1_STRIDE;
            v8f acc = {};
            #pragma unroll
            for (int kk = 0; kk < 32; ++kk) {
                int k0 = 4 * kk + 2 * hi;
                v2f b = *(const v2f*)&lds_w1[brow + k0];
                acc = __builtin_amdgcn_wmma_f32_16x16x4_f32(
                          false, a1[kk], false, b, (short)0, acc, false, false);
            }
            const float bv = bias1[g * HID + nloc + lm];
            #pragma unroll
            for (int v = 0; v < 8; ++v) {
                int mloc = v + 8 * hi;           // C layout: M = vgpr + 8*(lane/16)
                lds_h[(t2 * 16 + mloc) * H_STRIDE + (nloc + lm)] = gelu_exact(acc[v] + bv);
            }
        }
        __syncthreads();                          // H complete for both tiles

        // ---- layer 2: O_g(16x32 per wave) = H_g @ W2_g^T ----
        v8f acc2[2] = {v8f{}, v8f{}};
        const int hrow = (t2 * 16 + lm) * H_STRIDE;
        #pragma unroll 8
        for (int kk = 0; kk < 64; ++kk) {
            int k0 = 4 * kk + 2 * hi;
            v2f a = *(const v2f*)&lds_h[hrow + k0];
            #pragma unroll
            for (int nt2 = 0; nt2 < 2; ++nt2) {
                int brow = (nq * 32 + nt2 * 16 + lm) * W2_STRIDE;
                v2f b = *(const v2f*)&lds_w2[brow + k0];
                acc2[nt2] = __builtin_amdgcn_wmma_f32_16x16x4_f32(
                                false, a, false, b, (short)0, acc2[nt2], false, false);
            }
        }

        // ---- epilogue: out[m,4i+g] = O + b2[g*128+i] + res(m,i) ----
        #pragma unroll
        for (int nt2 = 0; nt2 < 2; ++nt2) {
            const int i = nq * 32 + nt2 * 16 + lm;       // 0..127
            const float b2 = bias2[g * SEG + i];
            const v4f wv = *(const v4f*)&w3d[4 * i];
            const float b3v = bias3[i];
            #pragma unroll
            for (int v = 0; v < 8; ++v) {
                int mloc = v + 8 * hi;
                long rowA = tileRow + mloc;
                v4f xv = *(const v4f*)&x[rowA * LS + 4 * i];
                float r = xv.x * wv.x + xv.y * wv.y + xv.z * wv.z + xv.w * wv.w + b3v;
                out[rowA * LS + 4 * i + g] = acc2[nt2][v] + b2 + r;
            }
        }
        __syncthreads();                          // protect lds_h for next chunk
    }
}

// ------------------------------------------------------------------
extern "C" void kernel_launch(void* const* d_in, const int* in_sizes, int n_in,
                              void* d_out, int out_size, void* d_ws, size_t ws_size,
                              hipStream_t stream) {
    const float* x    = (const float*)d_in[0];
    const float* wes1 = (const float*)d_in[1];
    const float* wes2 = (const float*)d_in[2];
    const float* wes3 = (const float*)d_in[3];
    const float* b1   = (const float*)d_in[4];
    const float* b2   = (const float*)d_in[5];
    const float* b3   = (const float*)d_in[6];
    float* out = (float*)d_out;

    float* w1p = (float*)d_ws;                 // 4*256*128
    float* w2p = w1p + 4 * 256 * 128;          // 4*128*256
    float* w3d = w2p + 4 * 128 * 256;          // 128*4

    const int M = in_sizes[0] / LS;            // bs*cs = 65536

    const int packN = 2 * 4 * 256 * 128 + SEG * SMP;
    maskmlp_pack_kernel<<<(packN + THREADS - 1) / THREADS, THREADS, 0, stream>>>(
        wes1, wes2, wes3, w1p, w2p, w3d);

    const int slabs = M / SLAB_ROWS;
    maskmlp_main_kernel<<<slabs * SMP, THREADS, 0, stream>>>(
        x, b1, b2, b3, w1p, w2p, w3d, out);
}